// HashEncoder_15401752723989
// MI455X (gfx1250) — compile-verified
//
#include <hip/hip_runtime.h>

// Instant-NGP hash-grid encoder for MI455X (gfx1250, wave32).
//
// Bound analysis: 67M independent 8B gathers into a 64MB table that is fully
// L2-resident (192MB global L2) -> L2-request-rate bound. HBM floor is only
// ~134MB (~6us @ 23.3TB/s). Structure:
//   * tiny precompute kernel writes the 16 per-level scales (float32
//     expf/logf/powf, matching XLA's b**arange semantics) into d_ws once per
//     launch -> hot kernel has NO transcendentals, no LDS, no barrier
//   * hot kernel per thread: 1 b32 scale load (64B line, L0-resident),
//     1 b96 x load, 8 saddr global_load_b64 gathers (32-bit offsets, scalar
//     table base -> 1 VALU of address math per gather), 1 coalesced b64 store
//   * all 8 gathers issued before any consumption; weight math overlaps latency
// WMMA is deliberately not used: the blend is a batched dot product with a
// per-point feature operand (no shared matmul operand); f16 packing would only
// add conversion cost + precision loss to an L2-request-bound kernel.

namespace {

constexpr int      kL       = 16;          // levels
constexpr int      kT       = 1 << 19;     // table rows per level
constexpr unsigned kTMask   = kT - 1;      // T is 2^19 -> floored mod == mask
constexpr int      kNPoints = 1 << 19;
constexpr unsigned kP1      = 2654435761u; // PIS[1]
constexpr unsigned kP2      = 805459861u;  // PIS[2]

// Launch-invariant per-level resolutions, replicating reference float32 math:
// b = exp(log(256)/15); scale_l = floor(16 * b^l). One wave, once per launch.
__global__ void scales_kernel(float* __restrict__ scales) {
  const int l = threadIdx.x;
  if (l < kL) {
    const float b = expf(logf(256.0f) / 15.0f);
    scales[l] = floorf(16.0f * powf(b, (float)l));
  }
}

__global__ __launch_bounds__(256)
void hash_encode_kernel(const float*  __restrict__ x,
                        const float2* __restrict__ table,   // (L*T) float2
                        const float*  __restrict__ scales,  // (L) f32, in d_ws
                        float2*       __restrict__ out) {   // (N*L) float2
  const unsigned tid   = blockIdx.x * 256u + threadIdx.x;   // = point*16 + level
  const unsigned level = tid & (kL - 1u);
  const unsigned n     = tid >> 4;

  // 64B scale table: one vmem instruction per wave, L0 hit after first touch.
  const float scale = scales[level];

  // Point coordinates (12B; compiler merges to one b96, L0 serves 16x reuse).
  const float x0 = x[3u * n + 0u];
  const float x1 = x[3u * n + 1u];
  const float x2 = x[3u * n + 2u];

  const float p0 = x0 * scale, p1 = x1 * scale, p2 = x2 * scale;
  const float g0 = floorf(p0), g1 = floorf(p1), g2 = floorf(p2);
  const int   i0 = (int)g0,    i1 = (int)g1,    i2 = (int)g2;
  const float fr0 = p0 - g0,   fr1 = p1 - g1,   fr2 = p2 - g2;

  // Per-dimension hash contributions for offset 0/1 (wrapping uint32 math,
  // identical to reference int32 corner*PIS then xor). (i+1)*P = i*P + P.
  const unsigned hy0 = (unsigned)i1 * kP1;
  const unsigned hz0 = (unsigned)i2 * kP2;
  const unsigned hx[2] = { (unsigned)i0, (unsigned)i0 + 1u };
  const unsigned hy[2] = { hy0,          hy0 + kP1          };
  const unsigned hz[2] = { hz0,          hz0 + kP2          };

  // 32-bit element offsets into the full (L*T) table: level base + hashed row.
  const unsigned lbase = level << 19;

  // Issue all 8 gathers up front: 8 outstanding saddr global_load_b64.
  float2 f[8];
#pragma unroll
  for (int c = 0; c < 8; ++c) {
    const unsigned h = hx[(c >> 2) & 1] ^ hy[(c >> 1) & 1] ^ hz[c & 1];
    f[c] = table[lbase + (h & kTMask)];
  }

  // Trilinear weights, factored: 4 xy products, then *wz per corner.
  const float wx[2] = { 1.0f - fr0, fr0 };
  const float wy[2] = { 1.0f - fr1, fr1 };
  const float wz[2] = { 1.0f - fr2, fr2 };
  float wxy[4];
#pragma unroll
  for (int c = 0; c < 4; ++c) wxy[c] = wx[(c >> 1) & 1] * wy[c & 1];

  float a0 = 0.0f, a1 = 0.0f;
#pragma unroll
  for (int c = 0; c < 8; ++c) {
    const float w = wxy[c >> 1] * wz[c & 1];
    a0 = fmaf(w, f[c].x, a0);
    a1 = fmaf(w, f[c].y, a1);
  }

  // out[n, level*2 + {0,1}] == out2[tid]: adjacent lanes -> adjacent float2s,
  // fully coalesced 256B-per-wave stores, 32-bit offset + scalar base.
  out[tid] = make_float2(a0, a1);
}

} // namespace

extern "C" void kernel_launch(void* const* d_in, const int* in_sizes, int n_in,
                              void* d_out, int out_size, void* d_ws, size_t ws_size,
                              hipStream_t stream) {
  const float*  x      = (const float*)d_in[0];  // (N_POINTS, 3) f32
  const float2* table  = (const float2*)d_in[1]; // (L, T, F=2)   f32
  float2*       out    = (float2*)d_out;         // (N_POINTS, L, F=2) f32
  float*        scales = (float*)d_ws;           // 16 floats of scratch

  // 1) per-level scales (launch-invariant, rewritten every call: deterministic)
  scales_kernel<<<1, 32, 0, stream>>>(scales);

  // 2) hot kernel
  constexpr int kThreads = 256;                  // 8 waves / block (wave32)
  constexpr int kTotal   = kNPoints * kL;        // 8,388,608 threads
  const int     blocks   = kTotal / kThreads;    // exact: 32768 blocks

  hash_encode_kernel<<<blocks, kThreads, 0, stream>>>(x, table, scales, out);
}